// HazardIntegrator_26259430048690
// MI455X (gfx1250) — compile-verified
//
#include <hip/hip_runtime.h>

// ---------------------------------------------------------------------------
// HazardIntegrator for MI455X (gfx1250, wave32)
//
// Kernel 1: one wave per 64-row unit (2 rows/lane as float2).
//   - TDM (tensor_load_to_lds) stages each wave's 512-float log_hazard / dt
//     slices into LDS; s_wait_tensorcnt 0 before consumption.
//   - exp -> trapezoid slices -> wave32 shfl_up inclusive scan (segmented
//     cumsum) -> survival -> second trapezoid.
//   - unit_E 32-lane reduction offloaded to the matrix pipe:
//     4x chained V_WMMA_F32_16X16X4_F32 with A=ones (co-executes with the
//     v_exp_f32-heavy VALU stream; XDL pipe is otherwise idle).
//   - writes sorted survival to d_ws (16 MB, stays hot in the 192 MB L2),
//     and per-unit scalars (lane 31).
// Kernel 2: permutation gather; recomputes hazard = exp(log_hazard[j]) and
//   gathers survival[j] from L2-resident scratch. int4/float4 vectorized.
//
// Roofline: ~115 MB of traffic total -> ~5 us at 23.3 TB/s; 8M v_exp_f32 is
// negligible. Random 4B gathers land in the 192 MB L2 (arrays are 16 MB).
// ---------------------------------------------------------------------------

#if defined(__HIP_DEVICE_COMPILE__) && defined(__gfx1250__) && \
    __has_builtin(__builtin_amdgcn_tensor_load_to_lds)
#define USE_TDM 1
#else
#define USE_TDM 0
#endif

#if defined(__HIP_DEVICE_COMPILE__) && defined(__gfx1250__) && \
    __has_builtin(__builtin_amdgcn_wmma_f32_16x16x4_f32)
#define USE_WMMA_REDUCE 1
#else
#define USE_WMMA_REDUCE 0
#endif

#if USE_TDM
typedef unsigned int tdm_u32x4 __attribute__((ext_vector_type(4)));
typedef int          tdm_i32x8 __attribute__((ext_vector_type(8)));
typedef int          tdm_i32x4 __attribute__((ext_vector_type(4)));

// 1-D f32 tile: global -> LDS via the Tensor Data Mover.
// D# group0: count=1, lds_addr, 57-bit global_addr, type=2 ("image").
// D# group1: data_size=4B, tensor_dim0=tile_dim0=nelem, dim0 stride=nelem.
__device__ __forceinline__ void tdm_load_1d_f32(unsigned long long gaddr_bytes,
                                                unsigned lds_byte_off,
                                                unsigned nelem) {
  tdm_u32x4 g0;
  g0.x = 1u;                                            // count=1 (valid user D#)
  g0.y = lds_byte_off;                                  // lds_addr [63:32]
  g0.z = (unsigned)(gaddr_bytes & 0xFFFFFFFFull);       // global_addr lo
  g0.w = ((unsigned)((gaddr_bytes >> 32) & 0x01FFFFFFull)) | (2u << 30); // hi + type=2
  tdm_i32x8 g1;
  g1[0] = (int)(2u << 16);          // workgroup_mask=0 (not in cluster), data_size=2 (4B)
  g1[1] = (int)(nelem << 16);       // tensor_dim0[15:0] at bits [63:48]
  g1[2] = (int)(1u << 16);          // tensor_dim0 hi = 0; tensor_dim1 = 1
  g1[3] = (int)(nelem << 16);       // tensor_dim1 hi = 0; tile_dim0 at bits [127:112]
  g1[4] = 0;                        // tile_dim1 = 0 (unused), tile_dim2 = 0
  g1[5] = (int)nelem;               // tensor_dim0_stride lo
  g1[6] = 0;                        // stride hi / dim1_stride lo
  g1[7] = 0;                        // dim1_stride hi
  tdm_i32x4 gz4 = {0, 0, 0, 0};     // groups 2/3 unused (<=2D tensor)
#if __clang_major__ >= 23
  tdm_i32x8 gz8 = {0, 0, 0, 0, 0, 0, 0, 0};
  __builtin_amdgcn_tensor_load_to_lds(g0, g1, gz4, gz4, gz8, 0);  // 6-arg form
#else
  __builtin_amdgcn_tensor_load_to_lds(g0, g1, gz4, gz4, 0);       // 5-arg form
#endif
}
#endif

__device__ __forceinline__ void wait_tensorcnt0() {
#if defined(__HIP_DEVICE_COMPILE__) && defined(__gfx1250__)
#if __has_builtin(__builtin_amdgcn_s_wait_tensorcnt)
  __builtin_amdgcn_s_wait_tensorcnt(0);
#else
  asm volatile("s_wait_tensorcnt 0x0" ::: "memory");
#endif
#endif
}

#if USE_WMMA_REDUCE
typedef float v2f __attribute__((ext_vector_type(2)));
typedef float v8f __attribute__((ext_vector_type(8)));

// 32-lane sum on the matrix pipe: D = sum_c ones(16x4) x B_c(4x16), with
// B_c[k][n] = x[k' + 16*(n>=8)] so every D element holds its column-half's
// 16-lane sum; combine halves with one xor-8 shuffle. Layouts per ISA 7.12.2.
__device__ __forceinline__ float wave_sum32_wmma(float x, int lane) {
  const int nsel = ((lane & 15) >= 8) ? 16 : 0;  // columns 8-15 take high lanes
  const int base = (lane < 16) ? 0 : 2;          // B rows 4c+{0,1} vs 4c+{2,3}
  v2f ones; ones.x = 1.0f; ones.y = 1.0f;        // A = ones(16x4)
  v8f acc = {0.f, 0.f, 0.f, 0.f, 0.f, 0.f, 0.f, 0.f};
#pragma unroll
  for (int c = 0; c < 4; ++c) {
    const int s = 4 * c + base + nsel;
    v2f b;
    b.x = __shfl(x, s, 32);                      // ds_bpermute gather of E[j]
    b.y = __shfl(x, s + 1, 32);
    acc = __builtin_amdgcn_wmma_f32_16x16x4_f32(false, ones, false, b,
                                                (short)0, acc, false, false);
  }
  return acc[0] + __shfl_xor(acc[0], 8u, 32);    // low half + high half
}
#endif

// ---- Kernel 1: per-unit segmented trapezoid / scan / reduce ----------------
// 256 threads = 8 waves; block covers 64 contiguous units (8 units per wave).
__global__ void __launch_bounds__(256) hazard_unit_kernel(
    const float* __restrict__ log_hazard, const float* __restrict__ dt,
    float* __restrict__ out_lasth, float* __restrict__ out_Lambda,
    float* __restrict__ out_E, float* __restrict__ surv_sorted) {
  __shared__ float smem[8192];  // [0,4096): log_hazard tile, [4096,8192): dt tile

  const int lane  = threadIdx.x & 31;
  const int wave  = threadIdx.x >> 5;
  const int unit0 = blockIdx.x * 64 + wave * 8;   // first unit for this wave
  const int row0  = unit0 * 64;                   // == blockIdx.x*4096 + wave*512
  (void)row0;

#if USE_TDM
  {
    const int w = __builtin_amdgcn_readfirstlane(wave);
    const unsigned lds_lh = (unsigned)(w * 2048);           // 512 floats
    const unsigned lds_dt = (unsigned)(16384 + w * 2048);
    const unsigned long long slice =
        ((unsigned long long)blockIdx.x * 4096ull + (unsigned long long)w * 512ull) * 4ull;
    tdm_load_1d_f32((unsigned long long)(size_t)log_hazard + slice, lds_lh, 512u);
    tdm_load_1d_f32((unsigned long long)(size_t)dt + slice, lds_dt, 512u);
    wait_tensorcnt0();
    // TDM wrote LDS behind the compiler's back: pass &smem into the clobber
    // so the LDS object escapes and the loads below cannot fold to undef.
    void* smem_escape = (void*)smem;
    asm volatile("" : "+v"(smem_escape) :: "memory");
  }
#else
  for (int t = lane; t < 512; t += 32) {
    smem[wave * 512 + t]        = log_hazard[row0 + t];
    smem[4096 + wave * 512 + t] = dt[row0 + t];
  }
  __syncthreads();
#endif

  for (int k = 0; k < 8; ++k) {
    const int unit = unit0 + k;
    const float2 lh2 = *(const float2*)&smem[wave * 512 + k * 64 + lane * 2];
    const float2 dt2 = *(const float2*)&smem[4096 + wave * 512 + k * 64 + lane * 2];

    // hazard for the two rows this lane owns
    const float h0 = __expf(lh2.x);
    const float h1 = __expf(lh2.y);

    // trapezoid slices (row 0 of the unit is a group start -> slice 0)
    const float h1p = __shfl_up(h1, 1u, 32);
    const float t0  = (lane == 0) ? 0.0f : 0.5f * (h0 + h1p) * dt2.x;
    const float t1  = 0.5f * (h1 + h0) * dt2.y;

    // wave32 inclusive scan of per-lane pair sums -> segmented cumsum
    float p = t0 + t1;
#pragma unroll
    for (int off = 1; off < 32; off <<= 1) {
      const float n = __shfl_up(p, (unsigned)off, 32);
      if (lane >= off) p += n;
    }
    const float Lam1 = p;        // Lambda at row 2*lane+1
    const float Lam0 = p - t1;   // Lambda at row 2*lane

    const float s0 = __expf(-Lam0);
    const float s1 = __expf(-Lam1);

    // survival trapezoid + 32-lane reduction for unit_E
    const float s1p = __shfl_up(s1, 1u, 32);
    const float e0  = (lane == 0) ? 0.0f : 0.5f * (s0 + s1p) * dt2.x;
    const float e1  = 0.5f * (s1 + s0) * dt2.y;
#if USE_WMMA_REDUCE
    const float E = wave_sum32_wmma(e0 + e1, lane);  // matrix-pipe reduction
#else
    float E = e0 + e1;
#pragma unroll
    for (int off = 16; off >= 1; off >>= 1) E += __shfl_xor(E, (unsigned)off, 32);
#endif

    // sorted survival -> scratch (consumed by gather kernel out of L2)
    *(float2*)&surv_sorted[unit * 64 + lane * 2] = make_float2(s0, s1);

    if (lane == 31) {
      out_lasth[unit]  = h1;    // hazard at last row of the unit
      out_Lambda[unit] = Lam1;  // full integrated hazard
      out_E[unit]      = E;     // expected time
    }
  }
}

// ---- Kernel 2: apply unsort permutation (gathers hit the 192 MB L2) --------
__global__ void __launch_bounds__(256) hazard_gather_kernel(
    const float* __restrict__ log_hazard, const int* __restrict__ unsort,
    const float* __restrict__ surv_sorted, float* __restrict__ out_h,
    float* __restrict__ out_s) {
  const int t = blockIdx.x * blockDim.x + threadIdx.x;
  const int4 j = ((const int4*)unsort)[t];
  float4 h, s;
  h.x = __expf(log_hazard[j.x]);  s.x = surv_sorted[j.x];
  h.y = __expf(log_hazard[j.y]);  s.y = surv_sorted[j.y];
  h.z = __expf(log_hazard[j.z]);  s.z = surv_sorted[j.z];
  h.w = __expf(log_hazard[j.w]);  s.w = surv_sorted[j.w];
  ((float4*)out_h)[t] = h;
  ((float4*)out_s)[t] = s;
}

extern "C" void kernel_launch(void* const* d_in, const int* in_sizes, int n_in,
                              void* d_out, int out_size, void* d_ws, size_t ws_size,
                              hipStream_t stream) {
  const float* lh     = (const float*)d_in[0];
  const float* dtv    = (const float*)d_in[1];
  const int*   unsort = (const int*)d_in[3];
  const int n     = in_sizes[0];   // 4,194,304 rows
  const int units = in_sizes[4];   // 65,536 units (64 rows each, contiguous)

  float* out       = (float*)d_out;
  float* out_h     = out;                        // hazard[unsort]
  float* out_lasth = out + n;                    // unit_last_hazard
  float* out_Lam   = out + n + units;            // unit_integrated_hazard
  float* out_s     = out + n + 2 * units;        // survival[unsort]
  float* out_E     = out + 2 * n + 2 * units;    // unit_expected_time

  float* surv = (float*)d_ws;                    // 16 MB sorted survival

  hazard_unit_kernel<<<dim3(units / 64), dim3(256), 0, stream>>>(
      lh, dtv, out_lasth, out_Lam, out_E, surv);
  hazard_gather_kernel<<<dim3(n / 1024), dim3(256), 0, stream>>>(
      lh, unsort, surv, out_h, out_s);
}